// LUTLayer_89472758710428
// MI455X (gfx1250) — compile-verified
//
#include <hip/hip_runtime.h>
#include <stdint.h>

#define INPUT_SIZE 4096
#define OUTPUT_SIZE 4096
#define NSEL 6
#define BATCH 2048
#define BWORDS (BATCH / 32)   // 64 bit-plane row-groups
#define TPB 256

typedef uint32_t u32;
typedef int v4i __attribute__((ext_vector_type(4)));

#if defined(__gfx1250__) && __has_builtin(__builtin_amdgcn_global_load_async_to_lds_b128)
#define HAVE_ASYNC_LDS 1
#else
#define HAVE_ASYNC_LDS 0
#endif

#if defined(__AMDGCN__)
typedef __attribute__((address_space(1))) v4i g_v4i;   // global (AS1)
typedef __attribute__((address_space(3))) v4i l_v4i;   // LDS (AS3)
#endif

// ---------------------------------------------------------------------------
// Kernel 1: collapse each neuron's 64-entry LUT to a 64-bit sign mask.
// binary output only depends on (lut > 0); clip(-1,1) never changes the sign.
// ---------------------------------------------------------------------------
__global__ void pack_luts_kernel(const float* __restrict__ luts,
                                 u32* __restrict__ lutbits) {
    int j = blockIdx.x * blockDim.x + threadIdx.x;
    if (j >= OUTPUT_SIZE) return;
    const float* row = luts + (size_t)j * 64;
    u32 lo = 0u, hi = 0u;
#pragma unroll
    for (int a = 0; a < 32; ++a) lo |= (row[a] > 0.0f ? 1u : 0u) << a;
#pragma unroll
    for (int a = 0; a < 32; ++a) hi |= (row[32 + a] > 0.0f ? 1u : 0u) << a;
    lutbits[2 * j]     = lo;
    lutbits[2 * j + 1] = hi;
}

// ---------------------------------------------------------------------------
// Kernel 2: bit-plane pack x.  xbits[bw*4096 + i] bit r = (x[bw*32+r][i] > 0.5)
// Reads of x are fully coalesced (lanes sweep i).
// ---------------------------------------------------------------------------
__global__ void pack_x_kernel(const float* __restrict__ x,
                              u32* __restrict__ xbits) {
    int i  = blockIdx.x * blockDim.x + threadIdx.x;
    int bw = blockIdx.y;
    u32 w = 0u;
#pragma unroll 8
    for (int r = 0; r < 32; ++r) {
        float v = x[(size_t)(bw * 32 + r) * INPUT_SIZE + i];
        w |= (v > 0.5f ? 1u : 0u) << r;
    }
    xbits[(size_t)bw * INPUT_SIZE + i] = w;
}

// ---------------------------------------------------------------------------
// Kernel 3: main LUT forward.
// Block = 256 neurons x one 32-row batch group. 16KB bit-plane row is staged
// into LDS via async global->LDS (ASYNCcnt) when available; each thread then
// does 6 LDS gathers and produces 32 outputs with pure bit math.
// ---------------------------------------------------------------------------
__global__ __launch_bounds__(TPB) void lut_forward_kernel(
    const u32* __restrict__ xbits, const int* __restrict__ mapping,
    const u32* __restrict__ lutbits, float* __restrict__ out) {
    __shared__ u32 xb[INPUT_SIZE];  // 16 KB

    const int tid = threadIdx.x;
    const int bw  = blockIdx.y;
    const u32* src = xbits + (size_t)bw * INPUT_SIZE;
    const int j = blockIdx.x * TPB + tid;
    const int* mp = mapping + (size_t)j * NSEL;

    // confirmed gfx1250 path: global_prefetch_b8 (pull mapping row toward L2/L0)
    __builtin_prefetch(mp, 0, 0);

    // ---- stage 16 KB of bit-planes into LDS: 16 dwords per thread ----------
    const int base = tid * 16;  // dword index
#if HAVE_ASYNC_LDS
    {
        g_v4i* g = (g_v4i*)(u32*)(src + base);
        l_v4i* l = (l_v4i*)(&xb[base]);
        __builtin_amdgcn_global_load_async_to_lds_b128(g + 0, l + 0, 0, 0);
        __builtin_amdgcn_global_load_async_to_lds_b128(g + 1, l + 1, 0, 0);
        __builtin_amdgcn_global_load_async_to_lds_b128(g + 2, l + 2, 0, 0);
        __builtin_amdgcn_global_load_async_to_lds_b128(g + 3, l + 3, 0, 0);
#if __has_builtin(__builtin_amdgcn_s_wait_asynccnt)
        __builtin_amdgcn_s_wait_asynccnt(0);
#else
        asm volatile("s_wait_asynccnt 0x0" ::: "memory");
#endif
    }
#else
    {
        const uint4* s4 = (const uint4*)(src + base);
        uint4*       d4 = (uint4*)(&xb[base]);
#pragma unroll
        for (int k = 0; k < 4; ++k) d4[k] = s4[k];
    }
#endif
    __syncthreads();

    // ---- gather the 6 selected bit-plane words for this neuron -------------
    const u32 w0 = xb[mp[0]];
    const u32 w1 = xb[mp[1]];
    const u32 w2 = xb[mp[2]];
    const u32 w3 = xb[mp[3]];
    const u32 w4 = xb[mp[4]];
    const u32 w5 = xb[mp[5]];

    const u32 llo = lutbits[2 * j];
    const u32 lhi = lutbits[2 * j + 1];

    float* orow = out + (size_t)(bw * 32) * OUTPUT_SIZE + j;

    // ---- 32 batch rows: assemble 6-bit address, test LUT sign bit ----------
#pragma unroll
    for (int r = 0; r < 32; ++r) {
        u32 addr = ((w0 >> r) & 1u)
                 | (((w1 >> r) & 1u) << 1)
                 | (((w2 >> r) & 1u) << 2)
                 | (((w3 >> r) & 1u) << 3)
                 | (((w4 >> r) & 1u) << 4)
                 | (((w5 >> r) & 1u) << 5);
        u32 word = (addr & 32u) ? lhi : llo;
        u32 bit  = (word >> (addr & 31u)) & 1u;
        orow[(size_t)r * OUTPUT_SIZE] = (float)bit;  // coalesced across lanes
    }
}

// ---------------------------------------------------------------------------
extern "C" void kernel_launch(void* const* d_in, const int* in_sizes, int n_in,
                              void* d_out, int out_size, void* d_ws,
                              size_t ws_size, hipStream_t stream) {
    const float* x       = (const float*)d_in[0];
    const int*   mapping = (const int*)d_in[1];
    const float* luts    = (const float*)d_in[2];
    float*       out     = (float*)d_out;

    // workspace layout: xbits (1 MB) then lutbits (32 KB)
    u32* xbits   = (u32*)d_ws;
    u32* lutbits = (u32*)((char*)d_ws + (size_t)BWORDS * INPUT_SIZE * sizeof(u32));

    pack_luts_kernel<<<OUTPUT_SIZE / 256, 256, 0, stream>>>(luts, lutbits);
    pack_x_kernel<<<dim3(INPUT_SIZE / 256, BWORDS), 256, 0, stream>>>(x, xbits);
    lut_forward_kernel<<<dim3(OUTPUT_SIZE / TPB, BWORDS), TPB, 0, stream>>>(
        xbits, mapping, lutbits, out);
}